// AFNO2DLayer_28819230556568
// MI455X (gfx1250) — compile-verified
//
#include <hip/hip_runtime.h>
#include <hip/hip_bf16.h>
#include <math.h>

// ---------------------------------------------------------------------------
// AFNO2D layer for MI455X (gfx1250, wave32).
//   x[1,256,256,768] --rfft2(ortho)--> per-(ky,kx) block-diag complex MLP
//   (8 blocks of 96, relu, softshrink) --irfft2--> + residual.
// FRAC==1.0 => the reference's mode mask is identically 1 and is elided.
//
// Roofline: ~2.1GB of HBM traffic @ 23.3TB/s (~90-120us) vs ~45 GFLOP fp32
// -> bandwidth-bound. Therefore: fp32 WMMA (precision-preserving, compute is
// free), fused 2-layer MLP (hidden tile in LDS, saves a 406MB round trip),
// TDM async tile loads into LDS for the GEMM stage.
//
// Pipeline (2 x 203MB workspace buffers A,B in d_ws):
//   1. k_fft_rows    : x -> A       (rFFT along W, keep 129 bins, *1/16)
//   2. k_fft_cols<-1>: A -> B       (complex FFT along H, *1/16)
//   3. k_mlp         : B -> A       (fused complex block MLP, WMMA f32 + TDM)
//   4. k_fft_cols<+1>: A -> B       (inverse FFT along H, *1/16)
//   5. k_ifft_rows   : B (+x) -> out(conj-symmetric irFFT along W, *1/16,+bias)
// ---------------------------------------------------------------------------

typedef __attribute__((ext_vector_type(2))) float v2f;
typedef __attribute__((ext_vector_type(8))) float v8f;
typedef __attribute__((ext_vector_type(4))) unsigned int u32x4;
typedef __attribute__((ext_vector_type(4))) int i32x4;
typedef __attribute__((ext_vector_type(8))) int i32x8;

#define HID   768
#define NBLK  8
#define BLK   96
#define HDIM  256
#define WDIM  256
#define WC    129          // WDIM/2 + 1
#define LOCS  (HDIM * WC)  // 33024 frequency locations (flattened ky*WC+kx)
#define LAMBDA 0.01f
#define ORTHO  (1.0f / 16.0f)   // ortho norm: 1/sqrt(256) per axis per direction

// Interleaved (re,im) LDS tile row stride in floats: 2*768 + 6.
// stride % 64 == 6 makes the 32-lane A-fragment read pattern
// {M=0..15} x {K,K+2} hit 32 distinct LDS banks (16-way-conflict-free).
#define XSTR  (2 * HID + 6)

// D = A(16x4) * B(4x16) + C, fp32 WMMA (ISA V_WMMA_F32_16X16X4_F32).
__device__ __forceinline__ v8f wmma_f32(v2f a, v2f b, v8f c) {
  return __builtin_amdgcn_wmma_f32_16x16x4_f32(false, a, false, b, (short)0, c,
                                               false, false);
}

// ---------------------------------------------------------------------------
// 256-point radix-2 DIT FFT over LDS, 16 independent channels per workgroup.
// Data must be pre-loaded in bit-reversed order. 2048 butterflies/stage over
// 256 threads. SIGN=-1 forward, +1 inverse.
// ---------------------------------------------------------------------------
template <int SIGN>
__device__ __forceinline__ void fft256_lds(float2 (*d)[17], int tid) {
#pragma unroll 1
  for (int s = 1; s <= 8; ++s) {
    const int m = 1 << s, half = m >> 1;
    __syncthreads();
#pragma unroll 1
    for (int it = 0; it < 8; ++it) {
      int idx = it * 256 + tid;          // 0..2047
      int c   = idx & 15;                // channel
      int bf  = idx >> 4;                // butterfly id 0..127
      int j   = bf & (half - 1);
      int g   = bf >> (s - 1);
      int p   = g * m + j, q = p + half;
      float ang = (float)SIGN * 6.28318530717958647692f * (float)j / (float)m;
      float sn, cs;
      __sincosf(ang, &sn, &cs);
      float2 u = d[p][c], w = d[q][c];
      float2 t = make_float2(w.x * cs - w.y * sn, w.x * sn + w.y * cs);
      d[p][c] = make_float2(u.x + t.x, u.y + t.y);
      d[q][c] = make_float2(u.x - t.x, u.y - t.y);
    }
  }
  __syncthreads();
}

// -------- 1: rFFT along W ---------------------------------------------------
__global__ __launch_bounds__(256) void k_fft_rows(const float* __restrict__ x,
                                                  float2* __restrict__ out) {
  __shared__ float2 d[256][17];
  const int tid = threadIdx.x;
  const int h   = blockIdx.x;
  const int c0  = blockIdx.y * 16;
  const float* row = x + (size_t)h * WDIM * HID;
#pragma unroll 1
  for (int it = 0; it < 16; ++it) {
    int idx = it * 256 + tid;  // coalesced: c fastest
    int w = idx >> 4, c = idx & 15;
    int wr = __brev((unsigned)w) >> 24;
    d[wr][c] = make_float2(row[(size_t)w * HID + c0 + c], 0.0f);
  }
  fft256_lds<-1>(d, tid);
#pragma unroll 1
  for (int it = 0; it < 9; ++it) {
    int idx = it * 256 + tid;
    if (idx < WC * 16) {
      int k = idx >> 4, c = idx & 15;
      float2 v = d[k][c];
      out[((size_t)h * WC + k) * HID + c0 + c] =
          make_float2(v.x * ORTHO, v.y * ORTHO);
    }
  }
}

// -------- 2/4: complex FFT along H (SIGN=-1 fwd, +1 inv) --------------------
template <int SIGN>
__global__ __launch_bounds__(256) void k_fft_cols(const float2* __restrict__ in,
                                                  float2* __restrict__ out) {
  __shared__ float2 d[256][17];
  const int tid = threadIdx.x;
  const int wc  = blockIdx.x;
  const int c0  = blockIdx.y * 16;
#pragma unroll 1
  for (int it = 0; it < 16; ++it) {
    int idx = it * 256 + tid;
    int hh = idx >> 4, c = idx & 15;
    int hr = __brev((unsigned)hh) >> 24;
    d[hr][c] = in[((size_t)hh * WC + wc) * HID + c0 + c];
  }
  fft256_lds<SIGN>(d, tid);
#pragma unroll 1
  for (int it = 0; it < 16; ++it) {
    int idx = it * 256 + tid;
    int k = idx >> 4, c = idx & 15;
    float2 v = d[k][c];
    out[((size_t)k * WC + wc) * HID + c0 + c] =
        make_float2(v.x * ORTHO, v.y * ORTHO);
  }
}

// -------- 3: fused 2-layer complex block MLP (WMMA f32 16x16x4 + TDM) -------
// Grid: LOCS/16 workgroups of 256 threads (8 waves). Wave b handles channel
// block b for the workgroup's 16-location tile. The tile lives interleaved
// (re,im) in LDS: loaded by 16 TDM row-descriptors (2 per wave, 6KB each),
// then overwritten in place by the hidden activations (wave-private columns).
__global__ __launch_bounds__(256) void k_mlp(const float2* __restrict__ in,
                                             const float* __restrict__ w1,
                                             const float* __restrict__ b1,
                                             const float* __restrict__ w2,
                                             const float* __restrict__ b2,
                                             float2* __restrict__ outp) {
  __shared__ float xin[16][XSTR];      // ~96KB interleaved complex tile

  const int tid  = threadIdx.x;
  const int lane = tid & 31;
  const int b    = tid >> 5;           // wave id == channel block 0..7
  const int lm   = lane & 15;          // M (A,C/D) or N (B,C/D) index
  const int lk   = (lane >> 4) << 1;   // K sub-offset {0,2} for A/B fragments
  const int mrow = (lane >> 4) << 3;   // C/D row base {0,8}
  const int l0   = blockIdx.x * 16;    // first flattened freq location

  // Weights are 1.15MB total -> pull into L2 once per WG (cheap, L2 = 192MB).
  for (int i = tid * 64; i < 2 * NBLK * BLK * BLK; i += 256 * 64) {
    __builtin_prefetch(w1 + i, 0, 3);
    __builtin_prefetch(w2 + i, 0, 3);
  }

  const float2* src = in + (size_t)l0 * HID;

#if __has_builtin(__builtin_amdgcn_tensor_load_to_lds)
  // ---- Tensor Data Mover load: wave b issues rows 2b and 2b+1 -------------
  // D# per CDNA5 ISA: count=1, type=2 ("image"),
  // data_size=8B, 1-D tile of 768 elements (one location's 768 cfloats).
  // 6-arg builtin form (clang-23 lane): (v4u, v8i, v4i, v4i, v8i, cpol).
#pragma unroll
  for (int rr = 0; rr < 2; ++rr) {
    const int m = b * 2 + rr;
    const unsigned long long ga =
        (unsigned long long)(const void*)(src + (size_t)m * HID);
    u32x4 g0;
    g0.x = 1u;                                        // count=1, user D#
    g0.y = (unsigned)(unsigned long long)(const void*)&xin[m][0];  // lds_addr
    g0.z = (unsigned)(ga & 0xFFFFFFFFull);            // global_addr[31:0]
    g0.w = (unsigned)((ga >> 32) & 0x1FFFFFFull) | (2u << 30);  // [56:32]|type
    i32x8 g1 = {};
    g1[0] = (int)(3u << 16);            // data_size = 8B
    g1[1] = (int)((HID & 0xFFFFu) << 16);   // tensor_dim0[15:0] @ bits 63:48
    g1[2] = 0;                          // tensor_dim0 hi / tensor_dim1 lo
    g1[3] = (int)((unsigned)HID << 16); // tile_dim0 = 768 @ bits 127:112
    g1[4] = 0;                          // tile_dim1/2 unused
    g1[5] = (int)HID;                   // tensor_dim0_stride[31:0]
    g1[6] = 0;
    g1[7] = 0;
    i32x4 z4 = {};
    i32x8 z8 = {};
    __builtin_amdgcn_tensor_load_to_lds(g0, g1, z4, z4, z8, 0);
  }
  __builtin_amdgcn_s_wait_tensorcnt(0);  // this wave's 2 TDM ops complete
#else
  // Fallback: cooperative coalesced load, kept interleaved (re,im).
#pragma unroll 1
  for (int it = 0; it < 48; ++it) {
    int idx = it * 256 + tid;          // idx = m*768 + c
    float2 v = src[idx];
    xin[idx / HID][2 * (idx % HID) + 0] = v.x;
    xin[idx / HID][2 * (idx % HID) + 1] = v.y;
  }
#endif
  __syncthreads();

  const float* W1r = w1 + (size_t)(0 * NBLK + b) * BLK * BLK;  // [i][o]
  const float* W1i = w1 + (size_t)(1 * NBLK + b) * BLK * BLK;
  const float* B1r = b1 + (0 * NBLK + b) * BLK;
  const float* B1i = b1 + (1 * NBLK + b) * BLK;

  // ---- layer 1: hidden = relu(X * (W1r + i W1i) + b1), kept in registers --
  v8f hR[6], hI[6];
#pragma unroll
  for (int n = 0; n < 6; ++n) {
    const int n0 = n * 16;
    const float br0 = B1r[n0 + lm], bi0 = B1i[n0 + lm];
    v8f accr = {}, accm = {}, acci = {};
#pragma unroll
    for (int r = 0; r < 8; ++r) { accr[r] = br0; acci[r] = bi0; }
#pragma unroll 4
    for (int kk = 0; kk < 24; ++kk) {
      const int k0 = kk * 4 + lk;      // this lane's K pair {k0, k0+1}
      const int xb = 2 * (BLK * b + k0);
      v2f ar = {xin[lm][xb + 0], xin[lm][xb + 2]};
      v2f ai = {xin[lm][xb + 1], xin[lm][xb + 3]};
      v2f wr = {W1r[(size_t)k0 * BLK + n0 + lm],
                W1r[(size_t)(k0 + 1) * BLK + n0 + lm]};
      v2f wi = {W1i[(size_t)k0 * BLK + n0 + lm],
                W1i[(size_t)(k0 + 1) * BLK + n0 + lm]};
      accr = wmma_f32(ar, wr, accr);   // Xr*Wr
      accm = wmma_f32(ai, wi, accm);   // Xi*Wi (subtracted below; f32 WMMA
                                       // has no A/B negate modifier)
      acci = wmma_f32(ai, wr, acci);   // Xi*Wr
      acci = wmma_f32(ar, wi, acci);   // + Xr*Wi
    }
    v8f hr, hi;
#pragma unroll
    for (int r = 0; r < 8; ++r) {
      hr[r] = fmaxf(accr[r] - accm[r], 0.0f);
      hi[r] = fmaxf(acci[r], 0.0f);
    }
    hR[n] = hr;
    hI[n] = hi;
  }

  // Overwrite this block's own column slice with the hidden activations.
  // Disjoint columns per wave; same-wave DS ops are in-order -> no barrier.
#pragma unroll
  for (int n = 0; n < 6; ++n) {
#pragma unroll
    for (int r = 0; r < 8; ++r) {
      const int xb = 2 * (BLK * b + n * 16 + lm);
      xin[mrow + r][xb + 0] = hR[n][r];
      xin[mrow + r][xb + 1] = hI[n][r];
    }
  }

  const float* W2r = w2 + (size_t)(0 * NBLK + b) * BLK * BLK;
  const float* W2i = w2 + (size_t)(1 * NBLK + b) * BLK * BLK;
  const float* B2r = b2 + (0 * NBLK + b) * BLK;
  const float* B2i = b2 + (1 * NBLK + b) * BLK;
  float2* dst = outp + (size_t)l0 * HID;

  // ---- layer 2: out = softshrink(Hid * (W2r + i W2i) + b2) ---------------
#pragma unroll 1
  for (int n = 0; n < 6; ++n) {
    const int n0 = n * 16;
    const float br0 = B2r[n0 + lm], bi0 = B2i[n0 + lm];
    v8f accr = {}, accm = {}, acci = {};
#pragma unroll
    for (int r = 0; r < 8; ++r) { accr[r] = br0; acci[r] = bi0; }
#pragma unroll 4
    for (int kk = 0; kk < 24; ++kk) {
      const int k0 = kk * 4 + lk;
      const int xb = 2 * (BLK * b + k0);
      v2f ar = {xin[lm][xb + 0], xin[lm][xb + 2]};
      v2f ai = {xin[lm][xb + 1], xin[lm][xb + 3]};
      v2f wr = {W2r[(size_t)k0 * BLK + n0 + lm],
                W2r[(size_t)(k0 + 1) * BLK + n0 + lm]};
      v2f wi = {W2i[(size_t)k0 * BLK + n0 + lm],
                W2i[(size_t)(k0 + 1) * BLK + n0 + lm]};
      accr = wmma_f32(ar, wr, accr);
      accm = wmma_f32(ai, wi, accm);
      acci = wmma_f32(ai, wr, acci);
      acci = wmma_f32(ar, wi, acci);
    }
#pragma unroll
    for (int r = 0; r < 8; ++r) {
      float vr = accr[r] - accm[r];
      float vi = acci[r];
      vr = copysignf(fmaxf(fabsf(vr) - LAMBDA, 0.0f), vr);  // softshrink
      vi = copysignf(fmaxf(fabsf(vi) - LAMBDA, 0.0f), vi);
      dst[(size_t)(mrow + r) * HID + BLK * b + n0 + lm] = make_float2(vr, vi);
    }
  }
}

// -------- 5: inverse rFFT along W + residual --------------------------------
__global__ __launch_bounds__(256) void k_ifft_rows(const float2* __restrict__ in,
                                                   const float* __restrict__ x,
                                                   float* __restrict__ outp) {
  __shared__ float2 d[256][17];
  const int tid = threadIdx.x;
  const int h   = blockIdx.x;
  const int c0  = blockIdx.y * 16;
  // Rebuild full 256-bin spectrum from 129 via conjugate symmetry, writing
  // straight into bit-reversed slots.
#pragma unroll 1
  for (int it = 0; it < 9; ++it) {
    int idx = it * 256 + tid;
    if (idx < WC * 16) {
      int k = idx >> 4, c = idx & 15;
      float2 v = in[((size_t)h * WC + k) * HID + c0 + c];
      d[__brev((unsigned)k) >> 24][c] = v;
      if (k >= 1 && k <= 127)
        d[__brev((unsigned)(256 - k)) >> 24][c] = make_float2(v.x, -v.y);
    }
  }
  fft256_lds<1>(d, tid);
  const float* row  = x + (size_t)h * WDIM * HID;
  float*       orow = outp + (size_t)h * WDIM * HID;
#pragma unroll 1
  for (int it = 0; it < 16; ++it) {
    int idx = it * 256 + tid;
    int w = idx >> 4, c = idx & 15;
    orow[(size_t)w * HID + c0 + c] =
        d[w][c].x * ORTHO + row[(size_t)w * HID + c0 + c];
  }
}

// ---------------------------------------------------------------------------
extern "C" void kernel_launch(void* const* d_in, const int* in_sizes, int n_in,
                              void* d_out, int out_size, void* d_ws,
                              size_t ws_size, hipStream_t stream) {
  (void)in_sizes; (void)n_in; (void)out_size; (void)ws_size;
  const float* x  = (const float*)d_in[0];
  const float* w1 = (const float*)d_in[1];
  const float* b1 = (const float*)d_in[2];
  const float* w2 = (const float*)d_in[3];
  const float* b2 = (const float*)d_in[4];
  float* out = (float*)d_out;

  float2* bufA = (float2*)d_ws;                       // 256*129*768 cfloat
  float2* bufB = bufA + (size_t)LOCS * HID;           // second 203MB buffer

  k_fft_rows<<<dim3(HDIM, HID / 16), 256, 0, stream>>>(x, bufA);
  k_fft_cols<-1><<<dim3(WC, HID / 16), 256, 0, stream>>>(bufA, bufB);
  k_mlp<<<LOCS / 16, 256, 0, stream>>>(bufB, w1, b1, w2, b2, bufA);
  k_fft_cols<1><<<dim3(WC, HID / 16), 256, 0, stream>>>(bufA, bufB);
  k_ifft_rows<<<dim3(HDIM, HID / 16), 256, 0, stream>>>(bufB, x, out);
}